// FraudGAT_12575664242836
// MI455X (gfx1250) — compile-verified
//
#include <hip/hip_runtime.h>
#include <math.h>

// ---------------------------------------------------------------------------
// FraudGAT for MI455X (gfx1250): 2-layer GAT + linear classifier.
// Dense GEMMs use V_WMMA_F32_16X16X4_F32 (f32 in / f32 acc -> no precision
// loss vs the f32 reference) with the B panel staged in LDS (320KB/WGP) and
// 4 accumulators per wave to amortize fragment loads. Graph phases are
// atomic-based segment softmax and scatter-add; the full working set
// (~130MB) is L2-resident (192MB).
// ---------------------------------------------------------------------------

typedef float v2f __attribute__((ext_vector_type(2)));
typedef float v8f __attribute__((ext_vector_type(8)));

#define NN      50000
#define NE      800000
#define ETOT    (NE + NN)      // with self loops
#define FIN     128
#define HIDC    64

// ---------------------------------------------------------------------------
// WMMA f32 GEMM: C[M,Nc] = A[M,K] @ B[K,Nc]
// Block = 256 threads = 8 waves; block tile = 128(M) x 64(N).
// Wave computes 16(M) x 64(N) via 4 accumulators, A fragment reused 4x.
// B panel (K x 64) staged once per block in LDS, k-pairs interleaved so each
// lane's two K-values are one contiguous ds_load_b64.
// Requires K%4==0, Nc%64==0, M%16==0 (all true here).
// ---------------------------------------------------------------------------
__global__ __launch_bounds__(256)
void k_gemm_wmma(const float* __restrict__ A, const float* __restrict__ B,
                 float* __restrict__ C, int M, int K, int Nc) {
    extern __shared__ float ldsB[];               // K*64 floats
    const int wave    = threadIdx.x >> 5;
    const int lane    = threadIdx.x & 31;
    const int tile_n0 = blockIdx.x * 64;
    const int tile_m  = (blockIdx.y * 8 + wave) * 16;

    // Cooperative stage of B[k][tile_n0 + c] -> lds[(k>>1)*128 + c*2 + (k&1)]
    for (int idx = threadIdx.x; idx < K * 64; idx += 256) {
        const int k = idx >> 6, c = idx & 63;
        ldsB[(k >> 1) * 128 + c * 2 + (k & 1)] = B[(size_t)k * Nc + tile_n0 + c];
    }
    __syncthreads();

    if (tile_m >= M) return;        // wave-uniform; after the only barrier

    const int lc    = lane & 15;
    const int khalf = (lane >> 4) * 2;            // 0 or 2 (always even)
    const float* arow = A + (size_t)(tile_m + lc) * K;

    v8f acc[4] = {{}, {}, {}, {}};
    for (int k = 0; k < K; k += 4) {
        const int kk = k + khalf;                 // even
        // A fragment: lane holds A[row][kk], A[row][kk+1] (one global b64)
        const v2f a = *(const v2f*)(arow + kk);
        // B fragments for the 4 n-tiles: one ds_load_b64 each
        const float* bp = ldsB + (kk >> 1) * 128 + lc * 2;
        const v2f b0 = *(const v2f*)(bp);
        const v2f b1 = *(const v2f*)(bp + 32);
        const v2f b2 = *(const v2f*)(bp + 64);
        const v2f b3 = *(const v2f*)(bp + 96);
        acc[0] = __builtin_amdgcn_wmma_f32_16x16x4_f32(false, a, false, b0,
                                                       (short)0, acc[0], false, false);
        acc[1] = __builtin_amdgcn_wmma_f32_16x16x4_f32(false, a, false, b1,
                                                       (short)0, acc[1], false, false);
        acc[2] = __builtin_amdgcn_wmma_f32_16x16x4_f32(false, a, false, b2,
                                                       (short)0, acc[2], false, false);
        acc[3] = __builtin_amdgcn_wmma_f32_16x16x4_f32(false, a, false, b3,
                                                       (short)0, acc[3], false, false);
    }

    // C/D layout: VGPR v -> row (tile_m + (lane>=16 ? 8 : 0) + v)
    const int out_row = tile_m + (lane >> 4) * 8;
#pragma unroll
    for (int t = 0; t < 4; ++t) {
#pragma unroll
        for (int v = 0; v < 8; ++v)
            C[(size_t)(out_row + v) * Nc + tile_n0 + t * 16 + lc] = acc[t][v];
    }
}

// ---------------------------------------------------------------------------
// Attention scores: s_src[n,h] = <h[n,h,:], att_src[h,:]>, same for dst.
// ---------------------------------------------------------------------------
__global__ void k_scores(const float* __restrict__ hfeat,
                         const float* __restrict__ att_src,
                         const float* __restrict__ att_dst,
                         float* __restrict__ s_src, float* __restrict__ s_dst,
                         int H) {
    const int i = blockIdx.x * blockDim.x + threadIdx.x;   // i = n*H + h
    if (i >= NN * H) return;
    const int hd = i % H;
    const float4* hv = (const float4*)(hfeat + (size_t)i * HIDC);
    const float4* as = (const float4*)(att_src + (size_t)hd * HIDC);
    const float4* ad = (const float4*)(att_dst + (size_t)hd * HIDC);
    float ss = 0.f, sd = 0.f;
#pragma unroll
    for (int j = 0; j < HIDC / 4; ++j) {
        const float4 x = hv[j], a = as[j], d = ad[j];
        ss += x.x * a.x + x.y * a.y + x.z * a.z + x.w * a.w;
        sd += x.x * d.x + x.y * d.y + x.z * d.z + x.w * d.w;
    }
    s_src[i] = ss;
    s_dst[i] = sd;
}

// ---------------------------------------------------------------------------
// Helpers
// ---------------------------------------------------------------------------
__device__ __forceinline__ void edge_endpoints(const int* __restrict__ ei,
                                               int e, int& s, int& d) {
    if (e < NE) { s = ei[e]; d = ei[NE + e]; }
    else        { s = e - NE; d = e - NE; }   // self loop
}

__device__ __forceinline__ float lrelu02(float x) {
    return x > 0.f ? x : 0.2f * x;
}

// sign-aware float atomic max via integer atomics (init with -inf)
__device__ __forceinline__ void atomic_max_f32(float* addr, float val) {
    if (val >= 0.f)
        atomicMax((int*)addr, __float_as_int(val));
    else
        atomicMin((unsigned int*)addr, (unsigned int)__float_as_int(val));
}

__global__ void k_fill(float* __restrict__ p, float val, int n) {
    const int i = blockIdx.x * blockDim.x + threadIdx.x;
    if (i < n) p[i] = val;
}

__global__ void k_fill_bias(float* __restrict__ out, const float* __restrict__ bias,
                            int HC, int total) {
    const int i = blockIdx.x * blockDim.x + threadIdx.x;
    if (i < total) out[i] = bias[i % HC];
}

// ---------------------------------------------------------------------------
// Segment softmax pass 1: per-dst running max of leaky_relu scores
// ---------------------------------------------------------------------------
__global__ void k_edge_max(const int* __restrict__ ei,
                           const float* __restrict__ s_src,
                           const float* __restrict__ s_dst,
                           float* __restrict__ m, int H) {
    const int e = blockIdx.x * blockDim.x + threadIdx.x;
    if (e >= ETOT) return;
    int s, d;
    edge_endpoints(ei, e, s, d);
    for (int hd = 0; hd < H; ++hd) {
        const float sc = lrelu02(s_src[(size_t)s * H + hd] + s_dst[(size_t)d * H + hd]);
        atomic_max_f32(&m[(size_t)d * H + hd], sc);
    }
}

// ---------------------------------------------------------------------------
// Segment softmax pass 2: exp(score - max), store per-edge, accumulate sums
// ---------------------------------------------------------------------------
__global__ void k_edge_exp(const int* __restrict__ ei,
                           const float* __restrict__ s_src,
                           const float* __restrict__ s_dst,
                           const float* __restrict__ m,
                           float* __restrict__ eexp, float* __restrict__ sum,
                           int H) {
    const int e = blockIdx.x * blockDim.x + threadIdx.x;
    if (e >= ETOT) return;
    int s, d;
    edge_endpoints(ei, e, s, d);
    for (int hd = 0; hd < H; ++hd) {
        const float sc = lrelu02(s_src[(size_t)s * H + hd] + s_dst[(size_t)d * H + hd]);
        const float ex = __expf(sc - m[(size_t)d * H + hd]);
        eexp[(size_t)e * H + hd] = ex;
        atomicAdd(&sum[(size_t)d * H + hd], ex);
    }
}

// ---------------------------------------------------------------------------
// Weighted message scatter: out[dst,h,c] += h[src,h,c] * alpha[e,h]
// One block per edge, one thread per (head, channel). out pre-seeded w/ bias.
// ---------------------------------------------------------------------------
__global__ void k_edge_aggregate(const int* __restrict__ ei,
                                 const float* __restrict__ hfeat,
                                 const float* __restrict__ eexp,
                                 const float* __restrict__ sum,
                                 float* __restrict__ out, int H) {
    const int e = blockIdx.x;
    const int t = threadIdx.x;           // 0 .. H*HIDC-1
    int s, d;
    edge_endpoints(ei, e, s, d);
    const int hd = t >> 6;               // t / HIDC
    const float alpha = eexp[(size_t)e * H + hd] /
                        (sum[(size_t)d * H + hd] + 1e-16f);
    atomicAdd(&out[(size_t)d * H * HIDC + t],
              hfeat[(size_t)s * H * HIDC + t] * alpha);
}

__global__ void k_elu(float* __restrict__ p, int n) {
    const int i = blockIdx.x * blockDim.x + threadIdx.x;
    if (i < n) {
        const float x = p[i];
        p[i] = x > 0.f ? x : (expf(x) - 1.f);
    }
}

// ---------------------------------------------------------------------------
// Classifier: out[n,0:2] = h[n,:] @ Wc + bc  (64x2, trivial)
// ---------------------------------------------------------------------------
__global__ void k_classifier(const float* __restrict__ h,
                             const float* __restrict__ Wc,
                             const float* __restrict__ bc,
                             float* __restrict__ out) {
    const int n = blockIdx.x * blockDim.x + threadIdx.x;
    if (n >= NN) return;
    float o0 = bc[0], o1 = bc[1];
    const float* hp = h + (size_t)n * HIDC;
#pragma unroll
    for (int c = 0; c < HIDC; ++c) {
        const float v = hp[c];
        o0 += v * Wc[c * 2 + 0];
        o1 += v * Wc[c * 2 + 1];
    }
    out[n * 2 + 0] = o0;
    out[n * 2 + 1] = o1;
}

// ---------------------------------------------------------------------------
// Host-side launcher
// ---------------------------------------------------------------------------
extern "C" void kernel_launch(void* const* d_in, const int* in_sizes, int n_in,
                              void* d_out, int out_size, void* d_ws, size_t ws_size,
                              hipStream_t stream) {
    const float* x        = (const float*)d_in[0];
    const int*   ei       = (const int*)  d_in[1];
    const float* W1       = (const float*)d_in[2];
    const float* att_src1 = (const float*)d_in[3];
    const float* att_dst1 = (const float*)d_in[4];
    const float* b1       = (const float*)d_in[5];
    const float* W2       = (const float*)d_in[6];
    const float* att_src2 = (const float*)d_in[7];
    const float* att_dst2 = (const float*)d_in[8];
    const float* b2       = (const float*)d_in[9];
    const float* Wc       = (const float*)d_in[10];
    const float* bc       = (const float*)d_in[11];
    float* out = (float*)d_out;

    // Workspace layout (floats). h1 region is reused for h2/agg2 after layer 1.
    float* ws    = (float*)d_ws;
    float* h1    = ws;                          // NN*256 = 12.8M
    float* agg1  = ws + (size_t)12800000;       // NN*256 = 12.8M
    float* eexp  = ws + (size_t)25600000;       // ETOT*4 = 3.4M (reused layer2)
    float* aux   = ws + (size_t)29000000;
    float* ssrc1 = aux;                         // NN*4
    float* sdst1 = aux + 200000;
    float* m1    = aux + 400000;
    float* sum1  = aux + 600000;
    float* ssrc2 = aux + 800000;                // NN
    float* sdst2 = aux + 850000;
    float* m2    = aux + 900000;
    float* sum2  = aux + 950000;
    float* h2    = h1;                          // NN*64, h1 dead by then
    float* agg2  = h1 + (size_t)4000000;        // NN*64

    const int TPB = 256;
    const float NEG_INF = -INFINITY;
    const int MTILES = NN / 16;                 // 3125
    const dim3 gemm_blk(256);

    // ---------------- Layer 1 (H=4) ----------------
    k_gemm_wmma<<<dim3(256 / 64, (MTILES + 7) / 8), gemm_blk,
                  FIN * 64 * sizeof(float), stream>>>(x, W1, h1, NN, FIN, 256);
    k_scores<<<(NN * 4 + TPB - 1) / TPB, TPB, 0, stream>>>(h1, att_src1, att_dst1,
                                                           ssrc1, sdst1, 4);
    k_fill<<<(NN * 4 + TPB - 1) / TPB, TPB, 0, stream>>>(m1, NEG_INF, NN * 4);
    k_fill<<<(NN * 4 + TPB - 1) / TPB, TPB, 0, stream>>>(sum1, 0.f, NN * 4);
    k_edge_max<<<(ETOT + TPB - 1) / TPB, TPB, 0, stream>>>(ei, ssrc1, sdst1, m1, 4);
    k_edge_exp<<<(ETOT + TPB - 1) / TPB, TPB, 0, stream>>>(ei, ssrc1, sdst1, m1,
                                                           eexp, sum1, 4);
    k_fill_bias<<<(NN * 256 + TPB - 1) / TPB, TPB, 0, stream>>>(agg1, b1, 256, NN * 256);
    k_edge_aggregate<<<ETOT, 256, 0, stream>>>(ei, h1, eexp, sum1, agg1, 4);
    k_elu<<<(NN * 256 + TPB - 1) / TPB, TPB, 0, stream>>>(agg1, NN * 256);

    // ---------------- Layer 2 (H=1) ----------------
    k_gemm_wmma<<<dim3(1, (MTILES + 7) / 8), gemm_blk,
                  256 * 64 * sizeof(float), stream>>>(agg1, W2, h2, NN, 256, 64);
    k_scores<<<(NN + TPB - 1) / TPB, TPB, 0, stream>>>(h2, att_src2, att_dst2,
                                                       ssrc2, sdst2, 1);
    k_fill<<<(NN + TPB - 1) / TPB, TPB, 0, stream>>>(m2, NEG_INF, NN);
    k_fill<<<(NN + TPB - 1) / TPB, TPB, 0, stream>>>(sum2, 0.f, NN);
    k_edge_max<<<(ETOT + TPB - 1) / TPB, TPB, 0, stream>>>(ei, ssrc2, sdst2, m2, 1);
    k_edge_exp<<<(ETOT + TPB - 1) / TPB, TPB, 0, stream>>>(ei, ssrc2, sdst2, m2,
                                                           eexp, sum2, 1);
    k_fill_bias<<<(NN * 64 + TPB - 1) / TPB, TPB, 0, stream>>>(agg2, b2, 64, NN * 64);
    k_edge_aggregate<<<ETOT, 64, 0, stream>>>(ei, h2, eexp, sum2, agg2, 1);

    // ---------------- Classifier ----------------
    k_classifier<<<(NN + TPB - 1) / TPB, TPB, 0, stream>>>(agg2, Wc, bc, out);
}